// ATT2_40707700032106
// MI455X (gfx1250) — compile-verified
//
#include <hip/hip_runtime.h>
#include <hip/hip_bf16.h>
#include <math.h>

#define BSZ    64
#define NIT    100
#define NPAD   112          // 7 * 16 row tiles
#define DIM    64
#define SM     68           // LDS row stride (floats) for M: (j*68+k)%64 unique over 16 rows
#define SR     68           // LDS row stride for R

typedef float v2f __attribute__((ext_vector_type(2)));
typedef float v8f __attribute__((ext_vector_type(8)));

__device__ __forceinline__ float wave_sum(float v) {
    #pragma unroll
    for (int m = 16; m > 0; m >>= 1) v += __shfl_xor(v, m, 32);
    return v;
}
__device__ __forceinline__ float wave_max(float v) {
    #pragma unroll
    for (int m = 16; m > 0; m >>= 1) v = fmaxf(v, __shfl_xor(v, m, 32));
    return v;
}

// ---------------------------------------------------------------------------
// Prologue: per (b,n) row, compute LN(em) then P = W_A @ LN, Q = W_B @ LN.
// grid = B*N blocks, 64 threads.
// ---------------------------------------------------------------------------
__global__ void ATT2_prologue(const float* __restrict__ emb,
                              const float* __restrict__ ln_g,
                              const float* __restrict__ ln_b,
                              const float* __restrict__ w_ij,
                              float* __restrict__ Pbuf,
                              float* __restrict__ Qbuf) {
    __shared__ float s_red[64];
    __shared__ float s_ln[64];
    const int row = blockIdx.x;
    const int t   = threadIdx.x;

    float x = emb[(size_t)row * DIM + t];
    s_red[t] = x;
    __syncthreads();
    #pragma unroll
    for (int s = 32; s > 0; s >>= 1) { if (t < s) s_red[t] += s_red[t + s]; __syncthreads(); }
    float mu = s_red[0] * (1.0f / DIM);
    __syncthreads();
    float d = x - mu;
    s_red[t] = d * d;
    __syncthreads();
    #pragma unroll
    for (int s = 32; s > 0; s >>= 1) { if (t < s) s_red[t] += s_red[t + s]; __syncthreads(); }
    float var = s_red[0] * (1.0f / DIM);
    float rs  = rsqrtf(var + 1e-5f);
    s_ln[t] = d * rs * ln_g[t] + ln_b[t];
    __syncthreads();

    float pt = 0.f, qt = 0.f;
    const float* wr = w_ij + (size_t)t * (3 * DIM);   // row t of w_ij (d', k)
    #pragma unroll 8
    for (int k = 0; k < DIM; ++k) {
        float l = s_ln[k];
        pt = fmaf(wr[k],        l, pt);   // W_A block
        qt = fmaf(wr[DIM + k],  l, qt);   // W_B block
    }
    Pbuf[(size_t)row * DIM + t] = pt;
    Qbuf[(size_t)row * DIM + t] = qt;
}

// ---------------------------------------------------------------------------
// Main: one workgroup per (b,i).  256 threads = 8 waves (wave32).
// ---------------------------------------------------------------------------
__global__ void __launch_bounds__(256)
ATT2_main(const float* __restrict__ emb,
          const float* __restrict__ ln_g,
          const float* __restrict__ ln_b,
          const float* __restrict__ w_ij,
          const float* __restrict__ b_ij,
          const float* __restrict__ w_sum,
          const float* __restrict__ b_sum,
          const float* __restrict__ Pbuf,
          const float* __restrict__ Qbuf,
          float* __restrict__ out) {
    __shared__ float s_M[NPAD * SM];     // LN(em_i * em_j), A operand
    __shared__ float s_R[NPAD * SR];     // GEMM result
    __shared__ float s_emi[DIM], s_g[DIM], s_b[DIM];
    __shared__ float s_pi[DIM], s_bij[DIM], s_wsum[DIM];
    __shared__ float s_z[NPAD], s_alpha[NPAD];
    __shared__ float s_part[256];
    __shared__ float s_scal[2];

    const int tid  = threadIdx.x;
    const int lane = tid & 31;
    const int wv   = tid >> 5;
    const int bidx = blockIdx.x;
    const int b    = bidx / NIT;
    const int i    = bidx % NIT;

    if (tid < DIM) {
        s_emi[tid]  = emb[((size_t)b * NIT + i) * DIM + tid];
        s_g[tid]    = ln_g[tid];
        s_b[tid]    = ln_b[tid];
        s_pi[tid]   = Pbuf[((size_t)b * NIT + i) * DIM + tid];
        s_bij[tid]  = b_ij[tid];
        s_wsum[tid] = w_sum[tid];
    }
    __syncthreads();

    // ---- Stage 1: build M[j,k] = LN(em_i * em_j) into LDS (rows >=100 zero)
    for (int j = wv; j < NPAD; j += 8) {
        if (j < NIT) {
            const float* ej = emb + ((size_t)b * NIT + j) * DIM;
            float v0 = s_emi[lane]      * ej[lane];
            float v1 = s_emi[lane + 32] * ej[lane + 32];
            float mu = wave_sum(v0 + v1) * (1.0f / DIM);
            float d0 = v0 - mu, d1 = v1 - mu;
            float var = wave_sum(d0 * d0 + d1 * d1) * (1.0f / DIM);
            float rs  = rsqrtf(var + 1e-5f);
            s_M[j * SM + lane]      = d0 * rs * s_g[lane]      + s_b[lane];
            s_M[j * SM + lane + 32] = d1 * rs * s_g[lane + 32] + s_b[lane + 32];
        } else {
            s_M[j * SM + lane]      = 0.f;
            s_M[j * SM + lane + 32] = 0.f;
        }
    }
    __syncthreads();

    // ---- Stage 2: WMMA GEMM  R[j,d'] = sum_k M[j,k] * W_C[d',k]
    // wave -> column tile ct = wv&3 (d' range n0..n0+15); row tiles rt = (wv>>2), +2, +4, +6
    {
        const int ct = wv & 3;
        const int n0 = ct * 16;
        const int nn = n0 + (lane & 15);
        const int kb = 2 * (lane >> 4);          // lanes 0-15 -> K offset 0,1 ; lanes 16-31 -> 2,3
        // preload all 16 B fragments for this column tile straight from global (L2)
        v2f bf[16];
        #pragma unroll
        for (int ks = 0; ks < 16; ++ks) {
            const float* wp = w_ij + (size_t)nn * (3 * DIM) + 2 * DIM + ks * 4 + kb;
            bf[ks].x = wp[0];
            bf[ks].y = wp[1];
        }
        for (int rt = (wv >> 2); rt < 7; rt += 2) {
            const int j0 = rt * 16;
            v8f acc = {0.f, 0.f, 0.f, 0.f, 0.f, 0.f, 0.f, 0.f};
            const int arow = (j0 + (lane & 15)) * SM + kb;
            #pragma unroll
            for (int ks = 0; ks < 16; ++ks) {
                v2f a;
                a.x = s_M[arow + ks * 4];
                a.y = s_M[arow + ks * 4 + 1];
                acc = __builtin_amdgcn_wmma_f32_16x16x4_f32(
                          false, a, false, bf[ks], (short)0, acc, false, false);
            }
            // D layout: VGPR r, lanes 0-15 -> row j0+r, lanes 16-31 -> row j0+8+r; col = n0+lane%16
            const int rbase = j0 + 8 * (lane >> 4);
            const int col   = n0 + (lane & 15);
            #pragma unroll
            for (int r = 0; r < 8; ++r)
                s_R[(rbase + r) * SR + col] = acc[r];
        }
    }
    __syncthreads();

    // ---- Stage 3: scores z[j] = w_sum . tanh(P_i + Q_j + R_j + b_ij) + b_sum
    const float bsum = b_sum[0];
    for (int j = wv; j < NIT; j += 8) {
        const float* q = Qbuf + ((size_t)b * NIT + j) * DIM;
        float part = 0.f;
        #pragma unroll
        for (int rr = 0; rr < 2; ++rr) {
            int dd = lane + rr * 32;
            float t = tanhf(s_pi[dd] + q[dd] + s_R[j * SR + dd] + s_bij[dd]);
            part = fmaf(s_wsum[dd], t, part);
        }
        part = wave_sum(part);
        if (lane == 0) s_z[j] = part + bsum;
    }
    __syncthreads();

    // ---- Stage 4: masked softmax over j
    if (wv == 0) {
        float m = -INFINITY;
        for (int j = lane; j < NIT; j += 32) {
            float z = (j == i) ? -INFINITY : s_z[j];
            m = fmaxf(m, z);
        }
        m = wave_max(m);
        float ssum = 0.f;
        for (int j = lane; j < NIT; j += 32) {
            float z = (j == i) ? -INFINITY : s_z[j];
            ssum += __expf(z - m);
        }
        ssum = wave_sum(ssum);
        if (lane == 0) { s_scal[0] = m; s_scal[1] = 1.0f / ssum; }
    }
    __syncthreads();
    if (tid < NPAD) {
        float a = 0.f;
        if (tid < NIT && tid != i) a = __expf(s_z[tid] - s_scal[0]) * s_scal[1];
        s_alpha[tid] = a;
    }
    __syncthreads();

    // ---- Stage 5: ctx_k = em_i[k] * sum_j alpha_j * em[b,j,k] + em_i[k]^2, leaky relu
    {
        const int k = tid & 63;
        const int g = tid >> 6;              // 4 groups of 25 j's
        float acc = 0.f;
        for (int j = g * 25; j < g * 25 + 25; ++j)
            acc = fmaf(s_alpha[j], emb[((size_t)b * NIT + j) * DIM + k], acc);
        s_part[tid] = acc;
    }
    __syncthreads();
    if (tid < DIM) {
        float ssum = s_part[tid] + s_part[64 + tid] + s_part[128 + tid] + s_part[192 + tid];
        float e = s_emi[tid];
        float c = fmaf(e, ssum, e * e);
        c = (c >= 0.f) ? c : 0.01f * c;
        out[((size_t)b * NIT + i) * DIM + tid] = c;
    }
}

// ---------------------------------------------------------------------------
extern "C" void kernel_launch(void* const* d_in, const int* in_sizes, int n_in,
                              void* d_out, int out_size, void* d_ws, size_t ws_size,
                              hipStream_t stream) {
    const float* emb   = (const float*)d_in[0];
    const float* ln_g  = (const float*)d_in[1];
    const float* ln_b  = (const float*)d_in[2];
    const float* w_ij  = (const float*)d_in[3];
    const float* b_ij  = (const float*)d_in[4];
    const float* w_sum = (const float*)d_in[5];
    const float* b_sum = (const float*)d_in[6];

    float* Pbuf = (float*)d_ws;
    float* Qbuf = Pbuf + (size_t)BSZ * NIT * DIM;

    dim3 gridP(BSZ * NIT), blockP(64);
    ATT2_prologue<<<gridP, blockP, 0, stream>>>(emb, ln_g, ln_b, w_ij, Pbuf, Qbuf);

    dim3 gridM(BSZ * NIT), blockM(256);
    ATT2_main<<<gridM, blockM, 0, stream>>>(emb, ln_g, ln_b, w_ij, b_ij, w_sum, b_sum,
                                            Pbuf, Qbuf, (float*)d_out);
}